// CRDLoss_41832981463421
// MI455X (gfx1250) — compile-verified
//
#include <hip/hip_runtime.h>
#include <math.h>

typedef float v2f __attribute__((ext_vector_type(2)));
typedef float v8f __attribute__((ext_vector_type(8)));

#define FEAT   128
#define SDIM   2048
#define BATCH  32
#define KNEG   8192
#define KP1    8193
#define TTEMP  0.07f
#define CRD_EPS 1e-7f

// ---------------------------------------------------------------------------
// K1: pre[side][m][n] = x[side] @ W[side]^T + bias, via V_WMMA_F32_16X16X4_F32.
// One wave per 16x16 output tile: 2 sides x 2 M-tiles x 8 N-tiles = 32 waves.
// f32 WMMA A layout: lanes 0-15 hold K=kk,kk+1; lanes 16-31 hold K=kk+2,kk+3.
// f32 WMMA B layout: lane = column n (0-15), VGPR0/1 = K rows {kk,kk+1} (low
// lanes) / {kk+2,kk+3} (high lanes)  -> both fragments are contiguous b64 loads.
// ---------------------------------------------------------------------------
__global__ __launch_bounds__(32)
void crd_embed_wmma(const float* __restrict__ x_s, const float* __restrict__ x_t,
                    const float* __restrict__ W_s, const float* __restrict__ W_t,
                    const float* __restrict__ b_s, const float* __restrict__ b_t,
                    float* __restrict__ pre)   // pre[2][BATCH][FEAT]
{
    int blk  = blockIdx.x;
    int side = blk >> 4;         // 0 = student, 1 = teacher
    int mt   = (blk >> 3) & 1;   // M tile (rows 0-15 / 16-31)
    int nt   = blk & 7;          // N tile (cols of the 128 features)

    const float* X    = side ? x_t : x_s;
    const float* W    = side ? W_t : W_s;
    const float* bias = side ? b_t : b_s;

    int lane = threadIdx.x;
    int lm   = lane & 15;
    int half = lane >> 4;
    int row  = mt * 16 + lm;     // sample index for A fragment
    int col  = nt * 16 + lm;     // feature index for B fragment

    const float* ap = X + (size_t)row * SDIM + 2 * half;
    const float* bp = W + (size_t)col * SDIM + 2 * half;

    v8f acc = {};
#pragma unroll 8
    for (int kk = 0; kk < SDIM; kk += 4) {
        v2f a = *(const v2f*)(ap + kk);
        v2f b = *(const v2f*)(bp + kk);
        acc = __builtin_amdgcn_wmma_f32_16x16x4_f32(false, a, false, b,
                                                    (short)0, acc, false, false);
    }

    // D layout: VGPR i -> row (i + 8*half), lane -> column lm.
    float bn = bias[col];
#pragma unroll
    for (int i = 0; i < 8; ++i) {
        int m = mt * 16 + i + 8 * half;
        pre[((size_t)side * BATCH + m) * FEAT + col] = acc[i] + bn;
    }
}

// ---------------------------------------------------------------------------
// K2: L2-normalize each of the 64 embedding rows (f = pre / ||pre||_2).
// ---------------------------------------------------------------------------
__global__ __launch_bounds__(128)
void crd_l2norm(const float* __restrict__ pre, float* __restrict__ f)
{
    int rowid = blockIdx.x;                 // 0..63 = side*32 + m
    int t     = threadIdx.x;                // 0..127 = feature
    __shared__ float red[4];

    float v  = pre[(size_t)rowid * FEAT + t];
    float ss = v * v;
#pragma unroll
    for (int off = 16; off > 0; off >>= 1)
        ss += __shfl_down(ss, off, 32);
    if ((t & 31) == 0) red[t >> 5] = ss;
    __syncthreads();
    float tot = red[0] + red[1] + red[2] + red[3];
    f[(size_t)rowid * FEAT + t] = v / sqrtf(tot);
}

// ---------------------------------------------------------------------------
// K3: the bandwidth-dominant gather-dot.
//   side 0: e[0][b][k] = exp(<memory_v2[row], f_s[b]>/T)   (out_v1 path)
//   side 1: e[1][b][k] = exp(<memory_v1[row], f_t[b]>/T)   (out_v2 path)
// One wave per 16-k tile; A = 16 gathered rows, B = f[b] broadcast across the
// 16 columns (so every D column is the same dot; we keep column 0).
// The per-(side) exp-sums for Z are accumulated via LDS + one global atomic.
// ---------------------------------------------------------------------------
__global__ __launch_bounds__(256)
void crd_gatherdot_wmma(const float* __restrict__ mem_v1,
                        const float* __restrict__ mem_v2,
                        const float* __restrict__ f,     // f[2][BATCH][FEAT]
                        const int*   __restrict__ idx,   // [BATCH]
                        const int*   __restrict__ cidx,  // [BATCH][KNEG]
                        float* __restrict__ e,           // e[2][BATCH][KP1]
                        float* __restrict__ gsum)        // [2]
{
    int blk   = blockIdx.x;
    int group = blk % 65;                 // 65 groups of 8 tiles cover 513 tiles
    int b     = (blk / 65) % BATCH;
    int side  = blk / (65 * BATCH);
    const float* mem = side ? mem_v1 : mem_v2;

    __shared__ float lds_f[FEAT];
    __shared__ float lds_acc;

    int tid = threadIdx.x;
    if (tid < FEAT)
        lds_f[tid] = f[((size_t)side * BATCH + b) * FEAT + tid];
    if (tid == 0) lds_acc = 0.0f;
    __syncthreads();

    int wave = tid >> 5;
    int lane = tid & 31;
    int lm   = lane & 15;
    int half = lane >> 4;
    int tile = group * 8 + wave;

    if (tile < 513) {                       // wave-uniform: EXEC stays all-1s
        int ki = tile * 16 + lm;
        int kc = ki > KNEG ? KNEG : ki;     // clamp tail (masked on store)
        int ridx = (kc == 0) ? idx[b] : cidx[(size_t)b * KNEG + (kc - 1)];
        const float* rowp = mem + (size_t)ridx * FEAT;
        __builtin_prefetch(rowp + half * 64, 0, 0);   // front-run the gather

        const float* ap = rowp + 2 * half;            // global b64 A fragments
        const float* bp = lds_f + 2 * half;           // LDS b64 B broadcast

        v8f acc = {};
#pragma unroll 8
        for (int kk = 0; kk < FEAT; kk += 4) {
            v2f a  = *(const v2f*)(ap + kk);
            v2f bb = *(const v2f*)(bp + kk);
            acc = __builtin_amdgcn_wmma_f32_16x16x4_f32(false, a, false, bb,
                                                        (short)0, acc, false, false);
        }

        if (lm == 0) {                       // column 0: lanes 0 (rows 0-7) and 16 (rows 8-15)
            float local = 0.0f;
#pragma unroll
            for (int i = 0; i < 8; ++i) {
                int kg = tile * 16 + i + 8 * half;
                if (kg <= KNEG) {
                    float ev = __expf(acc[i] * (1.0f / TTEMP));
                    e[((size_t)side * BATCH + b) * KP1 + kg] = ev;
                    local += ev;
                }
            }
            atomicAdd(&lds_acc, local);      // ds_add_f32
        }
    }
    __syncthreads();
    if (tid == 0) atomicAdd(&gsum[side], lds_acc);
}

// ---------------------------------------------------------------------------
// K4: NCE loss reduction over all 2*32*8193 exponentials -> scalar in d_out.
// Z_side = mean(e_side) * N_DATA; loss = -(sum of log terms)/BATCH.
// ---------------------------------------------------------------------------
__global__ __launch_bounds__(256)
void crd_loss_reduce(const float* __restrict__ e, const float* __restrict__ gsum,
                     float* __restrict__ out)
{
    const float c  = (float)KNEG / 1.0e6f;              // m * Pn
    const int total = 2 * BATCH * KP1;
    const float inv_mean = 1.0e6f / (float)(BATCH * KP1);
    float Zs = gsum[0] * inv_mean;
    float Zt = gsum[1] * inv_mean;

    float accum = 0.0f;
    for (int i = blockIdx.x * blockDim.x + threadIdx.x; i < total;
         i += gridDim.x * blockDim.x) {
        int side = i / (BATCH * KP1);
        int k    = i % KP1;
        float Z  = side ? Zt : Zs;
        float x  = e[i] / Z;
        float term = (k == 0) ? logf(x / (x + c + CRD_EPS))
                              : logf(c / (x + c + CRD_EPS));
        accum += term;
    }

    __shared__ float red[8];
#pragma unroll
    for (int off = 16; off > 0; off >>= 1)
        accum += __shfl_down(accum, off, 32);
    if ((threadIdx.x & 31) == 0) red[threadIdx.x >> 5] = accum;
    __syncthreads();
    if (threadIdx.x < 8) {
        float v = red[threadIdx.x];
#pragma unroll
        for (int off = 4; off > 0; off >>= 1)
            v += __shfl_down(v, off, 8);
        if (threadIdx.x == 0) atomicAdd(out, -v / (float)BATCH);
    }
}

// ---------------------------------------------------------------------------
extern "C" void kernel_launch(void* const* d_in, const int* in_sizes, int n_in,
                              void* d_out, int out_size, void* d_ws, size_t ws_size,
                              hipStream_t stream)
{
    const float* x_s = (const float*)d_in[0];
    const float* x_t = (const float*)d_in[1];
    const float* W_s = (const float*)d_in[2];
    const float* b_s = (const float*)d_in[3];
    const float* W_t = (const float*)d_in[4];
    const float* b_t = (const float*)d_in[5];
    const float* m1  = (const float*)d_in[6];
    const float* m2  = (const float*)d_in[7];
    const int*   idx  = (const int*)d_in[8];
    const int*   cidx = (const int*)d_in[9];
    float* out = (float*)d_out;

    // workspace layout (all 256B aligned): e | pre | f | gsum
    char*  ws     = (char*)d_ws;
    float* e      = (float*)ws;
    size_t eBytes = ((size_t)2 * BATCH * KP1 * sizeof(float) + 255) & ~(size_t)255;
    float* pre    = (float*)(ws + eBytes);
    float* f      = pre + (size_t)2 * BATCH * FEAT;
    float* gsum   = f   + (size_t)2 * BATCH * FEAT;

    hipMemsetAsync(gsum, 0, 2 * sizeof(float), stream);
    hipMemsetAsync(out,  0, sizeof(float), stream);

    crd_embed_wmma<<<32, 32, 0, stream>>>(x_s, x_t, W_s, W_t, b_s, b_t, pre);
    crd_l2norm<<<64, 128, 0, stream>>>(pre, f);
    crd_gatherdot_wmma<<<2 * BATCH * 65, 256, 0, stream>>>(m1, m2, f, idx, cidx, e, gsum);
    crd_loss_reduce<<<512, 256, 0, stream>>>(e, gsum, out);
}